// Temporal_Operator_14577119003475
// MI455X (gfx1250) — compile-verified
//
#include <hip/hip_runtime.h>
#include <cmath>

// out[b,t,c] = min over u in [t, min(T-1, t+63)] of in[b,u,c]
// B = 4096 rows, T = 4096, 2 interleaved channels (float2 per t).
// van Herk / Gil-Werman sliding-window min, window == block == 64.

#define T_LEN 4096
#define WIN 64
#define BLOCKS_PER_ROW (T_LEN / WIN)             // 64 blocks per row
#define ROWS_PER_WG 2
#define NTHREADS (ROWS_PER_WG * BLOCKS_PER_ROW)  // 128 threads = 4 wave32
#define PAD_STRIDE 65                            // float2 per padded 64-elem block (bank-conflict-free)
#define ROW_F2 (BLOCKS_PER_ROW * PAD_STRIDE)     // 4160 float2 per padded row

#if defined(__AMDGCN__) && __has_builtin(__builtin_amdgcn_global_load_async_to_lds_b64) && \
    __has_builtin(__builtin_amdgcn_global_store_async_from_lds_b64) && \
    __has_builtin(__builtin_amdgcn_s_wait_asynccnt)
#define USE_ASYNC_LDS 1
#else
#define USE_ASYNC_LDS 0
#endif

// Builtin param types: pointer-to-int2-vector, global (AS1) and LDS (AS3) qualified.
typedef int v2i __attribute__((vector_size(2 * sizeof(int))));
typedef __attribute__((address_space(1))) v2i gas_v2i;  // global int2
typedef __attribute__((address_space(3))) v2i las_v2i;  // LDS int2

__device__ __forceinline__ int pidx(int t) { return ((t >> 6) * PAD_STRIDE) + (t & 63); }

__global__ void __launch_bounds__(NTHREADS)
temporal_minwin64_kernel(const float2* __restrict__ gin, float2* __restrict__ gout,
                         int nrows_total) {
    extern __shared__ float2 lds[];
    float2* xbuf = lds;                          // [ROWS_PER_WG * ROW_F2] staged input, later output
    float2* pbuf = lds + ROWS_PER_WG * ROW_F2;   // [ROWS_PER_WG * ROW_F2] per-block prefix minima

    const int tid  = threadIdx.x;
    const int row0 = blockIdx.x * ROWS_PER_WG;
    int nrows = nrows_total - row0;
    if (nrows > ROWS_PER_WG) nrows = ROWS_PER_WG;
    const int nelem = nrows * T_LEN;
    const float2* rin  = gin  + (size_t)row0 * T_LEN;
    float2*       rout = gout + (size_t)row0 * T_LEN;

    // ---- Stage global -> LDS (coalesced 8B/lane; async DMA when available) ----
    for (int c = tid; c < nelem; c += NTHREADS) {
        const int r = c >> 12;                 // T_LEN == 4096
        const int t = c & (T_LEN - 1);
        const int l = r * ROW_F2 + pidx(t);
#if USE_ASYNC_LDS
        __builtin_amdgcn_global_load_async_to_lds_b64((gas_v2i*)(rin + c),
                                                      (las_v2i*)(xbuf + l), 0, 0);
#else
        xbuf[l] = rin[c];
#endif
    }
#if USE_ASYNC_LDS
    __builtin_amdgcn_s_wait_asynccnt(0);
#endif
    __syncthreads();

    const int r    = tid >> 6;               // row within WG
    const int k    = tid & 63;               // 64-element block within row
    const int base = r * ROW_F2 + k * PAD_STRIDE;

    // ---- Pass A: per-block running prefix min -> pbuf ----
    if (r < nrows) {
        float2 p = xbuf[base];
        pbuf[base] = p;
#pragma unroll 8
        for (int i = 1; i < WIN; ++i) {
            float2 v = xbuf[base + i];
            p.x = fminf(p.x, v.x);
            p.y = fminf(p.y, v.y);
            pbuf[base + i] = p;
        }
    }
    __syncthreads();

    // ---- Pass B: suffix min within block, combine with next block's prefix ----
    // out[t] = min( min(x[t..blockEnd]),  P[t+63] )   (P term dropped when t+63 >= T)
    if (r < nrows) {
        const int tbase = k * WIN;
        float2 s = make_float2(__builtin_inff(), __builtin_inff());
#pragma unroll 8
        for (int i = WIN - 1; i >= 0; --i) {
            float2 v = xbuf[base + i];
            s.x = fminf(s.x, v.x);
            s.y = fminf(s.y, v.y);
            float2 o = s;
            const int tn = tbase + i + (WIN - 1);
            if (tn < T_LEN) {
                float2 q = pbuf[r * ROW_F2 + pidx(tn)];
                o.x = fminf(o.x, q.x);
                o.y = fminf(o.y, q.y);
            }
            xbuf[base + i] = o;   // in place: only this thread re-reads this block of xbuf
        }
    }
    __syncthreads();

    // ---- Stage LDS -> global (coalesced 8B/lane; async DMA when available) ----
    for (int c = tid; c < nelem; c += NTHREADS) {
        const int rr = c >> 12;
        const int t  = c & (T_LEN - 1);
        const int l  = rr * ROW_F2 + pidx(t);
#if USE_ASYNC_LDS
        __builtin_amdgcn_global_store_async_from_lds_b64((gas_v2i*)(rout + c),
                                                         (las_v2i*)(xbuf + l), 0, 0);
#else
        rout[c] = xbuf[l];
#endif
    }
#if USE_ASYNC_LDS
    __builtin_amdgcn_s_wait_asynccnt(0);
#endif
}

extern "C" void kernel_launch(void* const* d_in, const int* in_sizes, int n_in,
                              void* d_out, int out_size, void* d_ws, size_t ws_size,
                              hipStream_t stream) {
    const float2* in  = (const float2*)d_in[0];
    float2*       out = (float2*)d_out;
    const int B = in_sizes[0] / (T_LEN * 2);                  // 4096
    const int grid = (B + ROWS_PER_WG - 1) / ROWS_PER_WG;     // 2048 workgroups
    const size_t shmem = (size_t)2 * ROWS_PER_WG * ROW_F2 * sizeof(float2);  // 133,120 B
    temporal_minwin64_kernel<<<dim3(grid), dim3(NTHREADS), shmem, stream>>>(in, out, B);
}